// MSDeformAttnTransformerEncoderOnly_5557687681304
// MI455X (gfx1250) — compile-verified
//
#include <hip/hip_runtime.h>
#include <hip/hip_bf16.h>

// ---------------------------------------------------------------------------
// Deformable-DETR encoder (MSDeformAttn), 6 layers, fp32, CDNA5 (gfx1250).
// GEMMs use V_WMMA_F32_16X16X4_F32 (fp32 WMMA, wave32).
// One wave computes a 64x32 C block (8 accumulators): per K-step of 4 we do
// 4 A-fragment b64 loads + 2 packed-B b64 loads + 8 independent WMMAs.
// N and K are template constants so all in-loop offsets fold into the
// 24-bit instruction IOFFSET (uniform saddr + one 32-bit lane offset).
// ---------------------------------------------------------------------------

typedef __attribute__((ext_vector_type(2))) float v2f;
typedef __attribute__((ext_vector_type(8))) float v8f;

#define NLAYERS 6
#define BD      4           // batch
#define DMODEL  256
#define NHEADS  8
#define HDIM    32
#define DFFN    1024
#define STOK    5440        // 64*64 + 32*32 + 16*16 + 8*8
#define MROWS   (BD * STOK) // 21760 = 340*64

// ---------------------------------------------------------------------------
// Repack W[K][N] -> Wp[(k>>1)*N + n][2] = {W[k][n], W[k+1][n]} so a B
// fragment (K rows kb, kb+1 at column col) is one 8B load. Pairs never
// straddle a layer boundary (every layer has an even row count).
// ---------------------------------------------------------------------------
__global__ __launch_bounds__(256) void repack_w(const float* __restrict__ W,
                                                float* __restrict__ Wp,
                                                int N, long total /* (K/2)*N */) {
  const long i = (long)blockIdx.x * 256 + threadIdx.x;
  if (i >= total) return;
  const long k2 = i / N;
  const long n  = i - k2 * N;
  const float w0 = W[(2 * k2) * (long)N + n];
  const float w1 = W[(2 * k2 + 1) * (long)N + n];
  float2 p; p.x = w0; p.y = w1;
  *(float2*)(Wp + i * 2) = p;
}

// q = a + b (elementwise, float4)
__global__ __launch_bounds__(256) void add_vec(const float* __restrict__ a,
                                               const float* __restrict__ b,
                                               float* __restrict__ q,
                                               long total4) {
  const long i = (long)blockIdx.x * 256 + threadIdx.x;
  if (i >= total4) return;
  const float4 av = ((const float4*)a)[i];
  const float4 bv = ((const float4*)b)[i];
  float4 r;
  r.x = av.x + bv.x; r.y = av.y + bv.y;
  r.z = av.z + bv.z; r.w = av.w + bv.w;
  ((float4*)q)[i] = r;
}

// ---------------------------------------------------------------------------
// GEMM: C[M,N] = A @ W + bias  (optional ReLU). W in packed K-pair layout.
// One wave -> 64x32 C block (A tiles i=0..3, B tiles j=0..1).
//   g = blockIdx.x*8 + wave;  mTile = g / (N/32);  nTile = g % (N/32)
// Fragment layouts per CDNA5 ISA 7.12.2: lanes 0-15 carry K={k,k+1},
// lanes 16-31 carry K={k+2,k+3}.
// ---------------------------------------------------------------------------
template <int N, int K, bool RELU>
__global__ __launch_bounds__(256) void gemm_wmma_f32(
    const float* __restrict__ A, const float* __restrict__ Wp,
    const float* __restrict__ bias, float* __restrict__ C, int M) {
  const int lane = threadIdx.x & 31;
  const int wave = threadIdx.x >> 5;
  const int hl   = lane >> 4;      // K-half of the wave
  const int lr   = lane & 15;      // row/col within a 16-wide stripe

  constexpr int nTiles = N >> 5;                  // 32-wide column tiles
  const int g      = blockIdx.x * 8 + wave;       // global wave id
  const int totalW = (M >> 6) * nTiles;
  if (g >= totalW) return;                        // wave-uniform
  const int m0 = (g / nTiles) << 6;               // 64-row block
  const int n0 = (g % nTiles) << 5;               // 32-col block

  const int col0 = n0 + lr;

  // lane-varying 32-bit element offsets; bases stay uniform (saddr form)
  const int aBase = (m0 + lr) * K + 2 * hl;       // A element offset
  const int bBase = (hl * N + col0) * 2;          // Wp element offset

  v8f acc00 = {}, acc01 = {}, acc10 = {}, acc11 = {};
  v8f acc20 = {}, acc21 = {}, acc30 = {}, acc31 = {};

#pragma unroll 4
  for (int k = 0; k < K; k += 4) {
    // A fragments: one b64 load each (compile-time tile offsets)
    const float2 av0 = *(const float2*)(A + aBase + 0 * 16 * K + k);
    const float2 av1 = *(const float2*)(A + aBase + 1 * 16 * K + k);
    const float2 av2 = *(const float2*)(A + aBase + 2 * 16 * K + k);
    const float2 av3 = *(const float2*)(A + aBase + 3 * 16 * K + k);
    // B fragments: packed K-pair -> one b64 load each
    const float2 bv0 = *(const float2*)(Wp + bBase + (k >> 1) * 2 * N + 0);
    const float2 bv1 = *(const float2*)(Wp + bBase + (k >> 1) * 2 * N + 32);
    v2f a0, a1, a2, a3, b0, b1;
    a0.x = av0.x; a0.y = av0.y;  a1.x = av1.x; a1.y = av1.y;
    a2.x = av2.x; a2.y = av2.y;  a3.x = av3.x; a3.y = av3.y;
    b0.x = bv0.x; b0.y = bv0.y;  b1.x = bv1.x; b1.y = bv1.y;
    // 8 independent WMMAs per K-step
    acc00 = __builtin_amdgcn_wmma_f32_16x16x4_f32(false, a0, false, b0, (short)0, acc00, false, false);
    acc01 = __builtin_amdgcn_wmma_f32_16x16x4_f32(false, a0, false, b1, (short)0, acc01, false, false);
    acc10 = __builtin_amdgcn_wmma_f32_16x16x4_f32(false, a1, false, b0, (short)0, acc10, false, false);
    acc11 = __builtin_amdgcn_wmma_f32_16x16x4_f32(false, a1, false, b1, (short)0, acc11, false, false);
    acc20 = __builtin_amdgcn_wmma_f32_16x16x4_f32(false, a2, false, b0, (short)0, acc20, false, false);
    acc21 = __builtin_amdgcn_wmma_f32_16x16x4_f32(false, a2, false, b1, (short)0, acc21, false, false);
    acc30 = __builtin_amdgcn_wmma_f32_16x16x4_f32(false, a3, false, b0, (short)0, acc30, false, false);
    acc31 = __builtin_amdgcn_wmma_f32_16x16x4_f32(false, a3, false, b1, (short)0, acc31, false, false);
  }

  const float bv0 = bias[col0];
  const float bv1 = bias[col0 + 16];
  const v8f* accs[4][2] = {{&acc00, &acc01}, {&acc10, &acc11},
                           {&acc20, &acc21}, {&acc30, &acc31}};
  const int cBase = (m0 + 8 * hl) * N + col0;     // lane-varying, bases uniform
#pragma unroll
  for (int i = 0; i < 4; ++i) {
#pragma unroll
    for (int v = 0; v < 8; ++v) {  // C tile: VGPR v holds M = v + 8*hl, N = lr
      const int r = cBase + (16 * i + v) * N;
      float e0 = (*accs[i][0])[v] + bv0;
      float e1 = (*accs[i][1])[v] + bv1;
      if (RELU) { e0 = fmaxf(e0, 0.0f); e1 = fmaxf(e1, 0.0f); }
      C[r] = e0;
      C[r + 16] = e1;
    }
  }
}

// ---------------------------------------------------------------------------
// Flatten (B,D,H,W) levels -> out(B,S,D) and pos(B,S,D)+level_embed.
// Also writes the int32 tuple tail (spatial_shapes, level_start_index).
// ---------------------------------------------------------------------------
__global__ __launch_bounds__(256) void flatten_levels(
    const float* __restrict__ s0, const float* __restrict__ s1,
    const float* __restrict__ s2, const float* __restrict__ s3,
    const float* __restrict__ p0, const float* __restrict__ p1,
    const float* __restrict__ p2, const float* __restrict__ p3,
    const float* __restrict__ le, float* __restrict__ out,
    float* __restrict__ pos, int* __restrict__ tail) {
  const size_t i = (size_t)blockIdx.x * 256 + threadIdx.x;
  const size_t total = (size_t)MROWS * DMODEL;
  if (i >= total) return;
  const int d = (int)(i & (DMODEL - 1));
  const size_t bs = i >> 8;
  const int s = (int)(bs % STOK);
  const int b = (int)(bs / STOK);
  const int st[4] = {0, 4096, 5120, 5376};
  const int nn[4] = {4096, 1024, 256, 64};
  int l = 3;
  if (s < 4096) l = 0; else if (s < 5120) l = 1; else if (s < 5376) l = 2;
  const int p = s - st[l];
  const float* sp[4] = {s0, s1, s2, s3};
  const float* pp[4] = {p0, p1, p2, p3};
  const size_t si = ((size_t)b * DMODEL + d) * nn[l] + p;
  out[i] = sp[l][si];
  pos[i] = pp[l][si] + le[l * DMODEL + d];
  if (i == 0) {
    const int vals[12] = {64, 64, 32, 32, 16, 16, 8, 8, 0, 4096, 5120, 5376};
#pragma unroll
    for (int t = 0; t < 12; ++t) tail[t] = vals[t];
  }
}

// ---------------------------------------------------------------------------
// MS deformable sampling: one wave32 per (b,s,h); lane = channel (HD==32).
// ---------------------------------------------------------------------------
__global__ __launch_bounds__(256) void msdeform_sample(
    const float* __restrict__ val, const float* __restrict__ so,
    const float* __restrict__ aw, float* __restrict__ att) {
  const int gw = (int)((blockIdx.x * 256u + threadIdx.x) >> 5);
  const int lane = threadIdx.x & 31;
  const int h = gw & (NHEADS - 1);
  const int bs = gw >> 3;
  if (bs >= MROWS) return;  // wave-uniform
  const int s = bs % STOK;
  const int b = bs / STOK;

  const int Wl[4] = {64, 32, 16, 8};
  const int Hl[4] = {64, 32, 16, 8};
  const int st[4] = {0, 4096, 5120, 5376};

  int ls = 3;
  if (s < 4096) ls = 0; else if (s < 5120) ls = 1; else if (s < 5376) ls = 2;
  const int local = s - st[ls];
  const float rx = ((local % Wl[ls]) + 0.5f) / (float)Wl[ls];
  const float ry = ((local / Wl[ls]) + 0.5f) / (float)Hl[ls];

  const float* lg = aw + (((size_t)bs * NHEADS) + h) * 16;
  float e[16];
  float mx = -1e30f;
#pragma unroll
  for (int j = 0; j < 16; ++j) { e[j] = lg[j]; mx = fmaxf(mx, e[j]); }
  float den = 0.0f;
#pragma unroll
  for (int j = 0; j < 16; ++j) { e[j] = __expf(e[j] - mx); den += e[j]; }
  const float rden = 1.0f / den;

  const float* off = so + (((size_t)bs * NHEADS) + h) * 32;
  const float* vb  = val + (size_t)b * STOK * DMODEL + h * HDIM + lane;

  float acc = 0.0f;
#pragma unroll
  for (int l = 0; l < 4; ++l) {
    const int Wc = Wl[l], Hc = Hl[l], s0 = st[l];
    const float rW = 1.0f / (float)Wc, rH = 1.0f / (float)Hc;
#pragma unroll
    for (int p = 0; p < 4; ++p) {
      const int j = l * 4 + p;
      const float a = e[j] * rden;
      const float lx = rx + off[j * 2 + 0] * rW;
      const float ly = ry + off[j * 2 + 1] * rH;
      const float x = lx * (float)Wc - 0.5f;
      const float y = ly * (float)Hc - 0.5f;
      const float x0f = floorf(x), y0f = floorf(y);
      const int x0 = (int)x0f, y0 = (int)y0f;
      const float wx1 = x - x0f, wy1 = y - y0f;
      const float wx0 = 1.0f - wx1, wy0 = 1.0f - wy1;
      const float cw[4] = {wx0 * wy0, wx1 * wy0, wx0 * wy1, wx1 * wy1};
#pragma unroll
      for (int c = 0; c < 4; ++c) {
        const int xi = x0 + (c & 1);
        const int yi = y0 + (c >> 1);
        const bool ok = (xi >= 0) & (xi < Wc) & (yi >= 0) & (yi < Hc);
        const float w = ok ? cw[c] * a : 0.0f;
        const int xc = min(max(xi, 0), Wc - 1);
        const int yc = min(max(yi, 0), Hc - 1);
        const int sidx = s0 + yc * Wc + xc;
        acc += w * vb[(size_t)sidx * DMODEL];
      }
    }
  }
  att[(((size_t)bs * NHEADS) + h) * HDIM + lane] = acc;
}

// ---------------------------------------------------------------------------
// out[row] = LayerNorm(base[row] + delta[row]) * g + be   (one block per row)
// ---------------------------------------------------------------------------
__global__ __launch_bounds__(256) void resid_layernorm(
    const float* __restrict__ base, const float* __restrict__ delta,
    const float* __restrict__ g, const float* __restrict__ be,
    float* __restrict__ out) {
  __shared__ float ssum[8], ssq[8], sm[2];
  const size_t rb = (size_t)blockIdx.x * DMODEL;
  const int d = threadIdx.x;
  const float v = base[rb + d] + delta[rb + d];
  float sum = v, sq = v * v;
#pragma unroll
  for (int o = 16; o > 0; o >>= 1) {
    sum += __shfl_xor(sum, o, 32);
    sq  += __shfl_xor(sq, o, 32);
  }
  if ((threadIdx.x & 31) == 0) {
    ssum[threadIdx.x >> 5] = sum;
    ssq[threadIdx.x >> 5]  = sq;
  }
  __syncthreads();
  if (threadIdx.x == 0) {
    float S = 0.0f, Q = 0.0f;
#pragma unroll
    for (int i = 0; i < 8; ++i) { S += ssum[i]; Q += ssq[i]; }
    const float mean = S * (1.0f / DMODEL);
    const float var  = Q * (1.0f / DMODEL) - mean * mean;
    sm[0] = mean;
    sm[1] = rsqrtf(var + 1e-5f);
  }
  __syncthreads();
  out[rb + d] = (v - sm[0]) * sm[1] * g[d] + be[d];
}

// ---------------------------------------------------------------------------
extern "C" void kernel_launch(void* const* d_in, const int* in_sizes, int n_in,
                              void* d_out, int out_size, void* d_ws,
                              size_t ws_size, hipStream_t stream) {
  const float* src[4] = {(const float*)d_in[0], (const float*)d_in[1],
                         (const float*)d_in[2], (const float*)d_in[3]};
  const float* psi[4] = {(const float*)d_in[4], (const float*)d_in[5],
                         (const float*)d_in[6], (const float*)d_in[7]};
  const float* le   = (const float*)d_in[8];
  const float* W_so = (const float*)d_in[9];
  const float* b_so = (const float*)d_in[10];
  const float* W_aw = (const float*)d_in[11];
  const float* b_aw = (const float*)d_in[12];
  const float* W_v  = (const float*)d_in[13];
  const float* b_v  = (const float*)d_in[14];
  const float* W_o  = (const float*)d_in[15];
  const float* b_o  = (const float*)d_in[16];
  const float* g1   = (const float*)d_in[17];
  const float* be1  = (const float*)d_in[18];
  const float* W_f1 = (const float*)d_in[19];
  const float* b_f1 = (const float*)d_in[20];
  const float* W_f2 = (const float*)d_in[21];
  const float* b_f2 = (const float*)d_in[22];
  const float* g2   = (const float*)d_in[23];
  const float* be2  = (const float*)d_in[24];

  float* out = (float*)d_out;                 // running activation (B,S,D)
  const size_t BSD = (size_t)MROWS * DMODEL;  // 5,570,560
  int* tail = (int*)(out + BSD);              // spatial_shapes + level_start

  // workspace carve-up (floats)
  float* pos = (float*)d_ws;                  // BSD
  float* q   = pos + BSD;                     // BSD
  float* so  = q + BSD;                       // BSD   (also reused as `t`)
  float* awb = so + BSD;                      // BSD/2
  float* val = awb + BSD / 2;                 // BSD
  float* att = val + BSD;                     // BSD
  float* x   = att + BSD;                     // BSD
  float* hb  = x + BSD;                       // 4*BSD (FFN hidden)
  float* t   = so;                            // reuse: so dead after sampling
  float* wsp = hb + 4 * BSD;                  // packed weights, 4,521,984 floats
  const size_t szSO = (size_t)NLAYERS * DMODEL * 256;
  const size_t szAW = (size_t)NLAYERS * DMODEL * 128;
  const size_t szVV = (size_t)NLAYERS * DMODEL * DMODEL;
  const size_t szF1 = (size_t)NLAYERS * DMODEL * DFFN;
  const size_t szF2 = (size_t)NLAYERS * DFFN * DMODEL;
  float* pSO = wsp;
  float* pAW = pSO + szSO;
  float* pVV = pAW + szAW;
  float* pOO = pVV + szVV;
  float* pF1 = pOO + szVV;
  float* pF2 = pF1 + szF1;

  const dim3 blk(256);
  const int nElemBlk = (int)((BSD + 255) / 256);
  const int n4Blk    = (int)((BSD / 4 + 255) / 256);

  // per-call weight repack: pairs of K-rows made contiguous
  {
    const long tSO = (long)szSO / 2, tAW = (long)szAW / 2, tVV = (long)szVV / 2;
    const long tF1 = (long)szF1 / 2, tF2 = (long)szF2 / 2;
    repack_w<<<(int)((tSO + 255) / 256), blk, 0, stream>>>(W_so, pSO, 256, tSO);
    repack_w<<<(int)((tAW + 255) / 256), blk, 0, stream>>>(W_aw, pAW, 128, tAW);
    repack_w<<<(int)((tVV + 255) / 256), blk, 0, stream>>>(W_v,  pVV, 256, tVV);
    repack_w<<<(int)((tVV + 255) / 256), blk, 0, stream>>>(W_o,  pOO, 256, tVV);
    repack_w<<<(int)((tF1 + 255) / 256), blk, 0, stream>>>(W_f1, pF1, 1024, tF1);
    repack_w<<<(int)((tF2 + 255) / 256), blk, 0, stream>>>(W_f2, pF2, 256, tF2);
  }

  flatten_levels<<<nElemBlk, blk, 0, stream>>>(
      src[0], src[1], src[2], src[3], psi[0], psi[1], psi[2], psi[3], le, out,
      pos, tail);

  // GEMM blocks: waves = (M/64)*(N/32), blocks = waves/8 (exact for all N)
  const int mT = MROWS / 64;                    // 340
  const int blkN128  = (mT * (128 / 32)) / 8;   // 170
  const int blkN256  = (mT * (256 / 32)) / 8;   // 340
  const int blkN1024 = (mT * (1024 / 32)) / 8;  // 1360

  const int waveBlocks = (MROWS * NHEADS) / 8;  // 21760 blocks of 8 waves
  const size_t lw = (size_t)DMODEL;             // per-layer vector stride

  for (int i = 0; i < NLAYERS; ++i) {
    // q = out + pos
    add_vec<<<n4Blk, blk, 0, stream>>>(out, pos, q, (long)(BSD / 4));
    // sampling offsets: so = q @ W_so + b_so    [N=256, K=256]
    gemm_wmma_f32<256, 256, false><<<blkN256, blk, 0, stream>>>(
        q, pSO + (size_t)i * DMODEL * 256, b_so + (size_t)i * 256, so, MROWS);
    // attention logits: awb = q @ W_aw + b_aw   [N=128, K=256]
    gemm_wmma_f32<128, 256, false><<<blkN128, blk, 0, stream>>>(
        q, pAW + (size_t)i * DMODEL * 128, b_aw + (size_t)i * 128, awb, MROWS);
    // value: val = out @ W_v + b_v              [N=256, K=256]
    gemm_wmma_f32<256, 256, false><<<blkN256, blk, 0, stream>>>(
        out, pVV + (size_t)i * DMODEL * DMODEL, b_v + i * lw, val, MROWS);
    // deformable sampling -> att
    msdeform_sample<<<waveBlocks, blk, 0, stream>>>(val, so, awb, att);
    // output projection: t = att @ W_o + b_o
    gemm_wmma_f32<256, 256, false><<<blkN256, blk, 0, stream>>>(
        att, pOO + (size_t)i * DMODEL * DMODEL, b_o + i * lw, t, MROWS);
    // x = LN(out + t)
    resid_layernorm<<<MROWS, blk, 0, stream>>>(out, t, g1 + i * lw,
                                               be1 + i * lw, x);
    // hb = relu(x @ W_f1 + b_f1)                [N=1024, K=256]
    gemm_wmma_f32<1024, 256, true><<<blkN1024, blk, 0, stream>>>(
        x, pF1 + (size_t)i * DMODEL * DFFN, b_f1 + (size_t)i * DFFN, hb, MROWS);
    // t = hb @ W_f2 + b_f2                      [N=256, K=1024]
    gemm_wmma_f32<256, 1024, false><<<blkN256, blk, 0, stream>>>(
        hb, pF2 + (size_t)i * DFFN * DMODEL, b_f2 + i * lw, t, MROWS);
    // out = LN(x + t)
    resid_layernorm<<<MROWS, blk, 0, stream>>>(x, t, g2 + i * lw,
                                               be2 + i * lw, out);
  }
}